// GraphAttentionLayer_68418829025697
// MI455X (gfx1250) — compile-verified
//
#include <hip/hip_runtime.h>
#include <hip/hip_bf16.h>
#include <math.h>

// GAT layer: N=8192 nodes, IN=OUT=512.
//   Wh = h@W              (WMMA f32 16x16x4)
//   f1 = Wh@a1, f2 = Wh@a2
//   e = leakyrelu(f1_i+f2_j) masked by adj, row softmax (online stats pass)
//   out = elu(P @ Wh)     (WMMA f32 16x16x4, P regenerated on the fly in LDS)

#define NN      8192
#define DIM     512
#define ALPHA   0.2f
#define NEG_BIG (-9.0e15f)

// k_wh LDS tile: 16 x 512 padded
#define WPAD    516
// k_attn LDS tile: 32 x 256 padded (stride 260 -> banks 4*m, conflict-free)
#define KCHUNK  256
#define PPAD    260
#define MBLK    32

typedef __attribute__((ext_vector_type(2))) float v2f;
typedef __attribute__((ext_vector_type(8))) float v8f;

static __device__ __forceinline__ v8f wmma_f32(v2f a, v2f b, v8f c) {
    return __builtin_amdgcn_wmma_f32_16x16x4_f32(
        /*neg_a=*/false, a, /*neg_b=*/false, b,
        /*c_mod=*/(short)0, c, /*reuse_a=*/false, /*reuse_b=*/false);
}

// ---------------------------------------------------------------------------
// Kernel 1: Wh = h @ W   (M=8192, K=512, N=512)  one 16-row strip per block
// ---------------------------------------------------------------------------
__global__ __launch_bounds__(256) void k_wh(const float* __restrict__ h,
                                            const float* __restrict__ W,
                                            float* __restrict__ Wh) {
    __shared__ float Ablk[16 * WPAD];
    const int i0   = blockIdx.x * 16;
    const int t    = threadIdx.x;
    const int wave = t >> 5;
    const int lane = t & 31;
    const int lm   = lane & 15;   // M (A) / N (B,C) index
    const int hi   = lane >> 4;   // half-wave selects K pair
    const int n0   = wave * 64;

    for (int idx = t; idx < 16 * DIM; idx += 256) {
        int r = idx >> 9, c = idx & 511;
        Ablk[r * WPAD + c] = h[(i0 + r) * DIM + c];
    }
    __syncthreads();

    v8f acc[4] = {};
    for (int k = 0; k < DIM; k += 4) {
        const int ka = k + hi * 2;
        v2f a;
        a.x = Ablk[lm * WPAD + ka];
        a.y = Ablk[lm * WPAD + ka + 1];
        const float* __restrict__ Bp = W + ka * DIM;
#pragma unroll
        for (int tt = 0; tt < 4; ++tt) {
            const int n = n0 + tt * 16 + lm;
            v2f b;
            b.x = Bp[n];
            b.y = Bp[DIM + n];
            acc[tt] = wmma_f32(a, b, acc[tt]);
        }
    }
#pragma unroll
    for (int tt = 0; tt < 4; ++tt)
#pragma unroll
        for (int v = 0; v < 8; ++v) {
            const int row = i0 + v + hi * 8;
            const int col = n0 + tt * 16 + lm;
            Wh[row * DIM + col] = acc[tt][v];
        }
}

// ---------------------------------------------------------------------------
// Kernel 2: f1[i] = Wh[i,:]·a1, f2[i] = Wh[i,:]·a2   (one wave per row)
// ---------------------------------------------------------------------------
__global__ __launch_bounds__(256) void k_f12(const float* __restrict__ Wh,
                                             const float* __restrict__ a,
                                             float* __restrict__ f1,
                                             float* __restrict__ f2) {
    const int wave = threadIdx.x >> 5, lane = threadIdx.x & 31;
    const int i = blockIdx.x * 8 + wave;
    float s1 = 0.f, s2 = 0.f;
    const float* __restrict__ wr = Wh + i * DIM;
    for (int c = lane; c < DIM; c += 32) {
        const float wv = wr[c];
        s1 += wv * a[c];
        s2 += wv * a[DIM + c];
    }
#pragma unroll
    for (int off = 16; off; off >>= 1) {
        s1 += __shfl_xor(s1, off, 32);
        s2 += __shfl_xor(s2, off, 32);
    }
    if (lane == 0) { f1[i] = s1; f2[i] = s2; }
}

// ---------------------------------------------------------------------------
// Kernel 3: per-row online softmax stats (max, 1/sum) over masked scores
//   masked entries carry NEG_BIG (NOT skipped) -> all-masked rows give
//   uniform attention exactly like jax softmax of a constant row.
// ---------------------------------------------------------------------------
__global__ __launch_bounds__(256) void k_stats(const int* __restrict__ adj,
                                               const float* __restrict__ f1,
                                               const float* __restrict__ f2,
                                               float* __restrict__ mrow,
                                               float* __restrict__ invl) {
    const int wave = threadIdx.x >> 5, lane = threadIdx.x & 31;
    const int i  = blockIdx.x * 8 + wave;
    const float fi = f1[i];
    const int* __restrict__ row = adj + (size_t)i * NN;
    float m = -INFINITY, s = 0.f;
    for (int j = lane; j < NN; j += 32) {
        __builtin_prefetch(row + j + 2048, 0, 0);   // global_prefetch_b8 (speculative)
        float e;
        if (row[j] > 0) {
            const float x = fi + f2[j];
            e = x > 0.f ? x : ALPHA * x;
        } else {
            e = NEG_BIG;
        }
        if (e <= m) {
            s += __expf(e - m);
        } else {
            s = s * __expf(m - e) + 1.f;
            m = e;
        }
    }
#pragma unroll
    for (int off = 16; off; off >>= 1) {
        const float om = __shfl_xor(m, off, 32);
        const float os = __shfl_xor(s, off, 32);
        if (om > m) {
            s = s * __expf(m - om) + os;
            m = om;
        } else {
            s += os * __expf(om - m);
        }
    }
    if (lane == 0) { mrow[i] = m; invl[i] = 1.f / s; }
}

// ---------------------------------------------------------------------------
// Kernel 4: out = elu(P @ Wh)   (M=8192, K=8192, N=512)
//   32-row block, K chunked by 256. All 256 threads expand the 32x256 P tile
//   into LDS once (adj read + expf); 8 waves each compute a 32x64 tile:
//   2 A frags (LDS) x 4 B frags (L2-resident Wh) -> 8 WMMAs per k-step,
//   i.e. 1 global dword per WMMA (2x better than a 16-row wave tile).
// ---------------------------------------------------------------------------
__global__ __launch_bounds__(256) void k_attn(const int* __restrict__ adj,
                                              const float* __restrict__ Wh,
                                              const float* __restrict__ f1,
                                              const float* __restrict__ f2,
                                              const float* __restrict__ mrow,
                                              const float* __restrict__ invl,
                                              float* __restrict__ out) {
    __shared__ float P[MBLK * PPAD];
    const int i0   = blockIdx.x * MBLK;
    const int t    = threadIdx.x;
    const int wave = t >> 5;
    const int lane = t & 31;
    const int lm   = lane & 15;
    const int hi   = lane >> 4;
    const int n0   = wave * 64;

    // P-fill assignment: thread t fills row pr (8 threads/row), 32 consecutive cols
    const int pr  = t >> 3;               // 0..31
    const int pc0 = (t & 7) * 32;         // 0..224
    const int gi  = i0 + pr;
    const float fi = f1[gi];
    const float mi = mrow[gi];
    const float li = invl[gi];
    const int* __restrict__ arow = adj + (size_t)gi * NN;

    v8f acc[2][4] = {};

    for (int kc = 0; kc < NN; kc += KCHUNK) {
        __syncthreads();   // previous chunk's readers done
        __builtin_prefetch(arow + kc + KCHUNK + pc0, 0, 0);  // next adj tile
#pragma unroll
        for (int u = 0; u < 32; u += 4) {
            const int c = pc0 + u;
            const int4   av  = *(const int4*)(arow + kc + c);
            const float4 fv  = *(const float4*)(f2 + kc + c);
            float e0, e1, e2, e3, x;
            x  = fi + fv.x; e0 = (av.x > 0) ? (x > 0.f ? x : ALPHA * x) : NEG_BIG;
            x  = fi + fv.y; e1 = (av.y > 0) ? (x > 0.f ? x : ALPHA * x) : NEG_BIG;
            x  = fi + fv.z; e2 = (av.z > 0) ? (x > 0.f ? x : ALPHA * x) : NEG_BIG;
            x  = fi + fv.w; e3 = (av.w > 0) ? (x > 0.f ? x : ALPHA * x) : NEG_BIG;
            float* __restrict__ pd = &P[pr * PPAD + c];
            pd[0] = __expf(e0 - mi) * li;
            pd[1] = __expf(e1 - mi) * li;
            pd[2] = __expf(e2 - mi) * li;
            pd[3] = __expf(e3 - mi) * li;
        }
        __syncthreads();   // P tile ready

        for (int k = 0; k < KCHUNK; k += 4) {
            const int ka = k + hi * 2;
            v2f a[2];
#pragma unroll
            for (int mt = 0; mt < 2; ++mt) {
                a[mt].x = P[(mt * 16 + lm) * PPAD + ka];
                a[mt].y = P[(mt * 16 + lm) * PPAD + ka + 1];
            }
            const float* __restrict__ Bp = Wh + (size_t)(kc + ka) * DIM;
            v2f b[4];
#pragma unroll
            for (int nt = 0; nt < 4; ++nt) {
                const int n = n0 + nt * 16 + lm;
                b[nt].x = Bp[n];
                b[nt].y = Bp[DIM + n];
            }
#pragma unroll
            for (int mt = 0; mt < 2; ++mt)
#pragma unroll
                for (int nt = 0; nt < 4; ++nt)
                    acc[mt][nt] = wmma_f32(a[mt], b[nt], acc[mt][nt]);
        }
    }

    // epilogue: elu + store
#pragma unroll
    for (int mt = 0; mt < 2; ++mt)
#pragma unroll
        for (int nt = 0; nt < 4; ++nt)
#pragma unroll
            for (int v = 0; v < 8; ++v) {
                const int row = i0 + mt * 16 + v + hi * 8;
                const int col = n0 + nt * 16 + lm;
                const float x = acc[mt][nt][v];
                out[row * DIM + col] = x > 0.f ? x : expm1f(x);
            }
}

// ---------------------------------------------------------------------------
extern "C" void kernel_launch(void* const* d_in, const int* in_sizes, int n_in,
                              void* d_out, int out_size, void* d_ws, size_t ws_size,
                              hipStream_t stream) {
    const float* h   = (const float*)d_in[0];  // 8192 x 512
    const int*   adj = (const int*)d_in[1];    // 8192 x 8192
    const float* W   = (const float*)d_in[2];  // 512 x 512
    const float* a   = (const float*)d_in[3];  // 1024 x 1
    float* out = (float*)d_out;                // 8192 x 512

    float* Wh   = (float*)d_ws;                // 16 MB
    float* f1   = Wh + (size_t)NN * DIM;
    float* f2   = f1 + NN;
    float* mrow = f2 + NN;
    float* invl = mrow + NN;

    k_wh   <<<NN / 16,   256, 0, stream>>>(h, W, Wh);
    k_f12  <<<NN / 8,    256, 0, stream>>>(Wh, a, f1, f2);
    k_stats<<<NN / 8,    256, 0, stream>>>(adj, f1, f2, mrow, invl);
    k_attn <<<NN / MBLK, 256, 0, stream>>>(adj, Wh, f1, f2, mrow, invl, out);
}